// VectorQuantizer_conv_47072841564924
// MI455X (gfx1250) — compile-verified
//
#include <hip/hip_runtime.h>

typedef __attribute__((ext_vector_type(2))) float v2f;
typedef __attribute__((ext_vector_type(8))) float v8f;

#define N_E    8192
#define E_DIM  256
#define N_Z    16384           // 8*2048
#define BETA   0.25f
#define LAMBDA_REG 0.1f
#define UNIFORM_WEIGHT 0.1f

// workspace layout (float index)
#define WS_EMBSQ 0             // sum(emb^2)
#define WS_LOSS  1             // sum((z_q - z)^2)
#define WS_UNI   2             // sum over rows of (max_d - min_d)
#define WS_ENORM 4             // [N_E] row norms ||e||^2
#define WS_IDX   (4 + N_E)     // [N_Z] int argmin indices

// output layout (floats, concatenated tuple)
#define OUT_IDX  (N_Z * E_DIM)      // 4194304
#define OUT_LOSS (OUT_IDX + N_Z)    // 4210688
#define OUT_QQ   (OUT_LOSS + 1)     // 4210689

__device__ __forceinline__ unsigned f2ord(float f) {
    unsigned b = __float_as_uint(f);
    return (b & 0x80000000u) ? ~b : (b | 0x80000000u);
}
__device__ __forceinline__ float ord2f(unsigned u) {
    unsigned b = (u & 0x80000000u) ? (u ^ 0x80000000u) : ~u;
    return __uint_as_float(b);
}

__global__ void vq_init_kernel(float* ws) {
    ws[WS_EMBSQ] = 0.f; ws[WS_LOSS] = 0.f; ws[WS_UNI] = 0.f;
}

// ||e_j||^2 per row + total sum(emb^2).  One wave per row, 8 rows / block.
__global__ void vq_enorm_kernel(const float* __restrict__ emb, float* __restrict__ ws) {
    const int lane = threadIdx.x & 31;
    const int wave = threadIdx.x >> 5;
    const int row  = blockIdx.x * 8 + wave;
    const float* p = emb + row * E_DIM + lane;
    float s = 0.f;
#pragma unroll
    for (int i = 0; i < E_DIM / 32; ++i) { float v = p[i * 32]; s += v * v; }
#pragma unroll
    for (int off = 16; off > 0; off >>= 1) s += __shfl_xor(s, off, 32);
    if (lane == 0) {
        ws[WS_ENORM + row] = s;
        atomicAdd(&ws[WS_EMBSQ], s);
    }
}

// argmin_j ( ||e_j||^2 - 2 z_i . e_j )   via fp32 WMMA.
// Block = 256 thr (8 waves) handles 16 z-rows; wave w covers codes [w*1024,(w+1)*1024).
__global__ void __launch_bounds__(256)
vq_argmin_kernel(const float* __restrict__ z, const float* __restrict__ emb,
                 const float* __restrict__ enorm, int* __restrict__ idx_out) {
    __shared__ unsigned long long best[16];
    const int lane = threadIdx.x & 31;
    const int wave = threadIdx.x >> 5;
    const int l15  = lane & 15;
    const int koff = 2 * (lane >> 4);                 // fp32 A/B fragment K sub-offset
    const int mrow = blockIdx.x * 16 + l15;

    // preload A fragments: z[mrow][k .. k+1] per K=4 chunk (ISA 16x4 fp32 layout)
    const float* zrow = z + mrow * E_DIM + koff;
    v2f afrag[E_DIM / 4];
#pragma unroll
    for (int kk = 0; kk < E_DIM / 4; ++kk)
        afrag[kk] = *(const v2f*)(zrow + 4 * kk);

    if (threadIdx.x < 16) best[threadIdx.x] = ~0ull;
    __syncthreads();

    unsigned long long run[8];
#pragma unroll
    for (int i = 0; i < 8; ++i) run[i] = ~0ull;

    for (int t = 0; t < 64; ++t) {
        const int nbase = wave * 1024 + t * 16;
        const float* erow = emb + (nbase + l15) * E_DIM + koff;
        v8f acc = {};
#pragma unroll
        for (int kk = 0; kk < E_DIM / 4; ++kk) {
            v2f b = *(const v2f*)(erow + 4 * kk);
            acc = __builtin_amdgcn_wmma_f32_16x16x4_f32(
                false, afrag[kk], false, b, (short)0, acc, false, false);
        }
        const int n  = nbase + l15;                   // N column this lane holds
        const float en = enorm[n];
#pragma unroll
        for (int i = 0; i < 8; ++i) {
            float val = en - 2.f * acc[i];            // ||z||^2 constant per row: dropped
            unsigned long long key =
                ((unsigned long long)f2ord(val) << 32) | (unsigned)n;
            run[i] = (key < run[i]) ? key : run[i];
        }
    }
    // slot i of this lane is row (i + 8*(lane>=16)) of the block's 16-row tile
#pragma unroll
    for (int i = 0; i < 8; ++i) {
        int mslot = i + ((lane < 16) ? 0 : 8);
        atomicMin(&best[mslot], run[i]);
    }
    __syncthreads();
    if (threadIdx.x < 16)
        idx_out[blockIdx.x * 16 + threadIdx.x] = (int)(best[threadIdx.x] & 0xFFFFFFFFu);
}

// cdist(emb,emb) row min/max -> accumulate sum(max-min), via fp32 WMMA.
__global__ void __launch_bounds__(256)
vq_cdist_kernel(const float* __restrict__ emb, const float* __restrict__ enorm,
                float* __restrict__ uni_accum) {
    __shared__ unsigned smin[16];
    __shared__ unsigned smax[16];
    const int lane = threadIdx.x & 31;
    const int wave = threadIdx.x >> 5;
    const int l15  = lane & 15;
    const int koff = 2 * (lane >> 4);
    const int mrow = blockIdx.x * 16 + l15;

    const float* arow = emb + mrow * E_DIM + koff;
    v2f afrag[E_DIM / 4];
#pragma unroll
    for (int kk = 0; kk < E_DIM / 4; ++kk)
        afrag[kk] = *(const v2f*)(arow + 4 * kk);

    // squared norms of the 8 M-rows this lane's acc slots correspond to
    float sqm[8];
#pragma unroll
    for (int i = 0; i < 8; ++i)
        sqm[i] = enorm[blockIdx.x * 16 + i + ((lane < 16) ? 0 : 8)];

    if (threadIdx.x < 16) { smin[threadIdx.x] = 0xFFFFFFFFu; smax[threadIdx.x] = 0u; }
    __syncthreads();

    unsigned rmin[8], rmax[8];
#pragma unroll
    for (int i = 0; i < 8; ++i) { rmin[i] = 0xFFFFFFFFu; rmax[i] = 0u; }

    for (int t = 0; t < 64; ++t) {
        const int nbase = wave * 1024 + t * 16;
        const float* erow = emb + (nbase + l15) * E_DIM + koff;
        v8f acc = {};
#pragma unroll
        for (int kk = 0; kk < E_DIM / 4; ++kk) {
            v2f b = *(const v2f*)(erow + 4 * kk);
            acc = __builtin_amdgcn_wmma_f32_16x16x4_f32(
                false, afrag[kk], false, b, (short)0, acc, false, false);
        }
        const float sqn = enorm[nbase + l15];
#pragma unroll
        for (int i = 0; i < 8; ++i) {
            float d2 = sqm[i] + sqn - 2.f * acc[i];
            d2 = fmaxf(d2, 0.f);
            float ed = (d2 > 0.f) ? sqrtf(d2) : 0.f;   // zero-safe sqrt as in reference
            unsigned u = f2ord(ed);
            rmin[i] = (u < rmin[i]) ? u : rmin[i];
            rmax[i] = (u > rmax[i]) ? u : rmax[i];
        }
    }
#pragma unroll
    for (int i = 0; i < 8; ++i) {
        int mslot = i + ((lane < 16) ? 0 : 8);
        atomicMin(&smin[mslot], rmin[i]);
        atomicMax(&smax[mslot], rmax[i]);
    }
    __syncthreads();
    if (threadIdx.x < 16) {
        float mn = ord2f(smin[threadIdx.x]);
        float mx = ord2f(smax[threadIdx.x]);
        atomicAdd(uni_accum, mx - mn);
    }
}

// gather z_q, emit z_q_st (== z_q numerically) + index-as-float + loss partial
__global__ void vq_gather_kernel(const float* __restrict__ z, const float* __restrict__ emb,
                                 const int* __restrict__ idx, float* __restrict__ out,
                                 float* __restrict__ ws) {
    __shared__ int sn;
    __shared__ float red[256];
    const int row = blockIdx.x;
    const int col = threadIdx.x;
    if (col == 0) sn = idx[row];
    __syncthreads();
    const int n = sn;
    float e  = emb[n * E_DIM + col];
    float zv = z[row * E_DIM + col];
    out[row * E_DIM + col] = e;                 // z + sg(z_q - z) == z_q
    float d = e - zv;
    red[col] = d * d;
    __syncthreads();
    for (int s = 128; s > 0; s >>= 1) {
        if (col < s) red[col] += red[col + s];
        __syncthreads();
    }
    if (col == 0) {
        atomicAdd(&ws[WS_LOSS], red[0]);
        out[OUT_IDX + row] = (float)n;
    }
}

__global__ void vq_finalize_kernel(const float* __restrict__ ws, float* __restrict__ out) {
    // loss = mean((sg(zq)-z)^2) + beta*mean((zq-sg(z))^2) = (1+beta)*mean((zq-z)^2)
    float loss = (1.f + BETA) * ws[WS_LOSS] / (float)(N_Z * E_DIM);
    float uniform_loss = ws[WS_UNI] / (float)N_E;
    float qq = UNIFORM_WEIGHT * uniform_loss + LAMBDA_REG * ws[WS_EMBSQ];
    out[OUT_LOSS] = loss;
    out[OUT_QQ]   = qq;
}

extern "C" void kernel_launch(void* const* d_in, const int* in_sizes, int n_in,
                              void* d_out, int out_size, void* d_ws, size_t ws_size,
                              hipStream_t stream) {
    const float* z   = (const float*)d_in[0];   // [8,2048,256]
    const float* emb = (const float*)d_in[1];   // [8192,256]
    float* out = (float*)d_out;
    float* ws  = (float*)d_ws;
    int*   idxbuf = ((int*)d_ws) + WS_IDX;

    vq_init_kernel<<<1, 1, 0, stream>>>(ws);
    vq_enorm_kernel<<<N_E / 8, 256, 0, stream>>>(emb, ws);
    vq_argmin_kernel<<<N_Z / 16, 256, 0, stream>>>(z, emb, ws + WS_ENORM, idxbuf);
    vq_cdist_kernel<<<N_E / 16, 256, 0, stream>>>(emb, ws + WS_ENORM, ws + WS_UNI);
    vq_gather_kernel<<<N_Z, 256, 0, stream>>>(z, emb, idxbuf, out, ws);
    vq_finalize_kernel<<<1, 1, 0, stream>>>(ws, out);
}